// GraphSAGE_87978110091549
// MI455X (gfx1250) — compile-verified
//
#include <hip/hip_runtime.h>
#include <math.h>

#define NN 40000      // N_NODES
#define NE 640000     // N_EDGES
#define INCH 128
#define HIDC 256
#define NG 64

typedef __attribute__((ext_vector_type(2))) float v2f;
typedef __attribute__((ext_vector_type(8))) float v8f;

// ---------------- CSR build ----------------

__global__ void zero_i32_kernel(int* __restrict__ p, int n) {
    int i = blockIdx.x * blockDim.x + threadIdx.x;
    if (i < n) p[i] = 0;
}

__global__ void count_deg_kernel(const int* __restrict__ ei, int* __restrict__ deg, int nE) {
    int e = blockIdx.x * blockDim.x + threadIdx.x;
    if (e < nE) atomicAdd(&deg[ei[nE + e]], 1);   // dst = ei[1][e]
}

// single-block exclusive scan over degrees; also emits inv_deg = 1/max(deg,1)
__global__ void scan_deg_kernel(const int* __restrict__ deg, int* __restrict__ row_ptr,
                                float* __restrict__ inv_deg, int n) {
    __shared__ int smem[1024];
    __shared__ int carry;
    if (threadIdx.x == 0) carry = 0;
    __syncthreads();
    for (int base = 0; base < n; base += 1024) {
        int i = base + (int)threadIdx.x;
        int v = (i < n) ? deg[i] : 0;
        smem[threadIdx.x] = v;
        __syncthreads();
        for (int off = 1; off < 1024; off <<= 1) {
            int t = (threadIdx.x >= (unsigned)off) ? smem[threadIdx.x - off] : 0;
            __syncthreads();
            smem[threadIdx.x] += t;
            __syncthreads();
        }
        int incl = smem[threadIdx.x];
        if (i < n) {
            row_ptr[i] = carry + (incl - v);
            inv_deg[i] = 1.0f / fmaxf((float)v, 1.0f);
        }
        __syncthreads();
        if (threadIdx.x == 1023) carry += smem[1023];
        __syncthreads();
    }
    if (threadIdx.x == 0) row_ptr[n] = carry;
}

__global__ void copy_i32_kernel(const int* __restrict__ a, int* __restrict__ b, int n) {
    int i = blockIdx.x * blockDim.x + threadIdx.x;
    if (i < n) b[i] = a[i];
}

__global__ void fill_csr_kernel(const int* __restrict__ ei, int* __restrict__ cursor,
                                int* __restrict__ csr_src, int nE) {
    int e = blockIdx.x * blockDim.x + threadIdx.x;
    if (e < nE) {
        int s = ei[e];            // src
        int d = ei[nE + e];       // dst
        int pos = atomicAdd(&cursor[d], 1);
        csr_src[pos] = s;
    }
}

// ---------------- mean aggregation: one wave32 per destination node ----------------

__global__ void aggregate_kernel(const float* __restrict__ h, const int* __restrict__ csr_src,
                                 const int* __restrict__ row_ptr, const float* __restrict__ inv_deg,
                                 float* __restrict__ agg, int din) {
    int wv   = (blockIdx.x * blockDim.x + threadIdx.x) >> 5;
    int lane = threadIdx.x & 31;
    if (wv >= NN) return;
    int beg = row_ptr[wv];
    int end = row_ptr[wv + 1];
    int nvec = din >> 7;  // float4 chunks per lane: 1 (din=128) or 2 (din=256)
    float4 a0 = make_float4(0.f, 0.f, 0.f, 0.f);
    float4 a1 = make_float4(0.f, 0.f, 0.f, 0.f);
    for (int e = beg; e < end; ++e) {
        int src = csr_src[e];
        if (e + 1 < end) {  // prefetch next neighbor row into cache (global_prefetch_b8)
            __builtin_prefetch(h + (size_t)csr_src[e + 1] * din, 0, 0);
        }
        const float4* hs = (const float4*)(h + (size_t)src * din);
        float4 v0 = hs[lane];
        a0.x += v0.x; a0.y += v0.y; a0.z += v0.z; a0.w += v0.w;
        if (nvec > 1) {
            float4 v1 = hs[lane + 32];
            a1.x += v1.x; a1.y += v1.y; a1.z += v1.z; a1.w += v1.w;
        }
    }
    float s = inv_deg[wv];
    float4* ao = (float4*)(agg + (size_t)wv * din);
    ao[lane] = make_float4(a0.x * s, a0.y * s, a0.z * s, a0.w * s);
    if (nvec > 1) ao[lane + 32] = make_float4(a1.x * s, a1.y * s, a1.z * s, a1.w * s);
}

// ---------------- SAGE GEMM: out = elu(agg@wl^T + bl + h@wr^T) ----------------
// One wave computes a 16x64 output strip (4 N-tiles) so each A load feeds 4 WMMAs.
// V_WMMA_F32_16X16X4_F32 operand layout (ISA 7.12.2):
//   A (16x4): lanes 0-15 M=lane, lanes 16-31 M=lane-16; VGPR j holds K = 2*half + j
//   B (4x16): N=lane%16;                                VGPR j holds K = 2*half + j
//   C/D (16x16): VGPR v holds M = v + 8*half, N = lane%16

template <int DIN>
__device__ __forceinline__ void wmma_kpass(const float* __restrict__ A, const float* __restrict__ W,
                                           int m0, int n0, int r, int hb, v8f c[4]) {
    const float* arow = A + (size_t)(m0 + r) * DIN + 2 * hb;
    const float* w0 = W + (size_t)(n0 + r) * DIN + 2 * hb;            // B[k][n] = W[n][k]
    const float* w1 = w0 + (size_t)16 * DIN;
    const float* w2 = w0 + (size_t)32 * DIN;
    const float* w3 = w0 + (size_t)48 * DIN;
#pragma unroll 8
    for (int k = 0; k < DIN; k += 4) {
        v2f a, b0, b1, b2, b3;
        a[0]  = arow[k];  a[1]  = arow[k + 1];
        b0[0] = w0[k];    b0[1] = w0[k + 1];
        b1[0] = w1[k];    b1[1] = w1[k + 1];
        b2[0] = w2[k];    b2[1] = w2[k + 1];
        b3[0] = w3[k];    b3[1] = w3[k + 1];
        c[0] = __builtin_amdgcn_wmma_f32_16x16x4_f32(false, a, false, b0, (short)0, c[0], false, false);
        c[1] = __builtin_amdgcn_wmma_f32_16x16x4_f32(false, a, false, b1, (short)0, c[1], false, false);
        c[2] = __builtin_amdgcn_wmma_f32_16x16x4_f32(false, a, false, b2, (short)0, c[2], false, false);
        c[3] = __builtin_amdgcn_wmma_f32_16x16x4_f32(false, a, false, b3, (short)0, c[3], false, false);
    }
}

template <int DIN>
__global__ __launch_bounds__(256) void sage_gemm_kernel(
        const float* __restrict__ agg, const float* __restrict__ h,
        const float* __restrict__ wl, const float* __restrict__ bl,
        const float* __restrict__ wr, float* __restrict__ out) {
    int wid  = (blockIdx.x * blockDim.x + threadIdx.x) >> 5;  // 2500 mtiles x 4 nstrips
    int lane = threadIdx.x & 31;
    int mtile  = wid >> 2;
    int nstrip = wid & 3;
    int m0 = mtile * 16;
    int n0 = nstrip * 64;
    int hb = lane >> 4;
    int r  = lane & 15;

    v8f c[4];
#pragma unroll
    for (int t = 0; t < 4; ++t) c[t] = (v8f){0.f, 0.f, 0.f, 0.f, 0.f, 0.f, 0.f, 0.f};

    wmma_kpass<DIN>(agg, wl, m0, n0, r, hb, c);   // lin_l(mean-agg)
    wmma_kpass<DIN>(h,   wr, m0, n0, r, hb, c);   // lin_r(root)

#pragma unroll
    for (int t = 0; t < 4; ++t) {
        float bias = bl[n0 + t * 16 + r];
#pragma unroll
        for (int v = 0; v < 8; ++v) {
            int m = m0 + v + 8 * hb;
            float val = c[t][v] + bias;
            val = (val > 0.f) ? val : (__expf(val) - 1.f);   // ELU, alpha=1
            out[(size_t)m * HIDC + n0 + t * 16 + r] = val;
        }
    }
}

// ---------------- readout: out[g] = head_b + sum_{n in g} dot(h[n], head_w) ----------------

__global__ void init_out_kernel(float* __restrict__ out, const float* __restrict__ head_b, int n) {
    int i = blockIdx.x * blockDim.x + threadIdx.x;
    if (i < n) out[i] = head_b[0];
}

__global__ void readout_kernel(const float* __restrict__ h, const int* __restrict__ batch,
                               const float* __restrict__ head_w, float* __restrict__ out) {
    int wv   = (blockIdx.x * blockDim.x + threadIdx.x) >> 5;
    int lane = threadIdx.x & 31;
    if (wv >= NN) return;
    const float* row = h + (size_t)wv * HIDC;
    float s = 0.f;
#pragma unroll
    for (int k = 0; k < HIDC / 32; ++k) s += row[lane + k * 32] * head_w[lane + k * 32];
    for (int off = 16; off > 0; off >>= 1) s += __shfl_xor(s, off, 32);
    if (lane == 0) atomicAdd(&out[batch[wv]], s);
}

// ---------------- launch ----------------

extern "C" void kernel_launch(void* const* d_in, const int* in_sizes, int n_in,
                              void* d_out, int out_size, void* d_ws, size_t ws_size,
                              hipStream_t stream) {
    const float* x      = (const float*)d_in[0];
    const int*   ei     = (const int*)d_in[1];
    const int*   batch  = (const int*)d_in[2];
    const float* wl[4]  = {(const float*)d_in[3],  (const float*)d_in[6],
                           (const float*)d_in[9],  (const float*)d_in[12]};
    const float* blv[4] = {(const float*)d_in[4],  (const float*)d_in[7],
                           (const float*)d_in[10], (const float*)d_in[13]};
    const float* wr[4]  = {(const float*)d_in[5],  (const float*)d_in[8],
                           (const float*)d_in[11], (const float*)d_in[14]};
    const float* head_w = (const float*)d_in[15];
    const float* head_b = (const float*)d_in[16];
    float* out = (float*)d_out;

    char* ws = (char*)d_ws;
    auto alloc = [&](size_t bytes) -> char* {
        char* p = ws;
        ws += (bytes + 255) & ~(size_t)255;
        return p;
    };
    int*   deg     = (int*)alloc((size_t)NN * 4);
    int*   row_ptr = (int*)alloc((size_t)(NN + 1) * 4);
    int*   cursor  = (int*)alloc((size_t)NN * 4);
    int*   csr_src = (int*)alloc((size_t)NE * 4);
    float* inv_deg = (float*)alloc((size_t)NN * 4);
    float* agg     = (float*)alloc((size_t)NN * HIDC * 4);
    float* h0      = (float*)alloc((size_t)NN * HIDC * 4);
    float* h1      = (float*)alloc((size_t)NN * HIDC * 4);

    // CSR build (recomputed every call -> deterministic)
    zero_i32_kernel<<<(NN + 255) / 256, 256, 0, stream>>>(deg, NN);
    count_deg_kernel<<<(NE + 255) / 256, 256, 0, stream>>>(ei, deg, NE);
    scan_deg_kernel<<<1, 1024, 0, stream>>>(deg, row_ptr, inv_deg, NN);
    copy_i32_kernel<<<(NN + 255) / 256, 256, 0, stream>>>(row_ptr, cursor, NN);
    fill_csr_kernel<<<(NE + 255) / 256, 256, 0, stream>>>(ei, cursor, csr_src, NE);

    const int AGG_BLOCKS  = (NN * 32) / 256;            // one wave per node, 8 waves/block
    const int GEMM_BLOCKS = ((NN / 16) * 4 * 32) / 256; // 2500 mtiles x 4 nstrips, 8 waves/block

    const float* hc = x;
    float* bufs[2] = {h0, h1};
    for (int l = 0; l < 4; ++l) {
        aggregate_kernel<<<AGG_BLOCKS, 256, 0, stream>>>(hc, csr_src, row_ptr, inv_deg, agg,
                                                         (l == 0) ? INCH : HIDC);
        float* hn = bufs[l & 1];
        if (l == 0) {
            sage_gemm_kernel<INCH><<<GEMM_BLOCKS, 256, 0, stream>>>(agg, hc, wl[l], blv[l], wr[l], hn);
        } else {
            sage_gemm_kernel<HIDC><<<GEMM_BLOCKS, 256, 0, stream>>>(agg, hc, wl[l], blv[l], wr[l], hn);
        }
        hc = hn;
    }

    init_out_kernel<<<1, 64, 0, stream>>>(out, head_b, NG);
    readout_kernel<<<AGG_BLOCKS, 256, 0, stream>>>(hc, batch, head_w, out);
}